// AlloLayer_60799557042737
// MI455X (gfx1250) — compile-verified
//
#include <hip/hip_runtime.h>
#include <hip/hip_bf16.h>
#include <stdint.h>

#define B_    32
#define T_    1500
#define C_    1024
#define L_    100
#define P_    256
#define S_    201
#define SPAD  224          // 32 lanes * 7 states each
#define EMSTR 256          // padded em row stride (floats) -> 1KB rows, async-b128 friendly
#define NEGF  (-1.0e30f)
#define CHROWS 8           // time-rows per async staging chunk

// ---- CDNA5 async-to-LDS probing (device pass only) -------------------------
#if defined(__HIP_DEVICE_COMPILE__) && __has_builtin(__builtin_amdgcn_global_load_async_to_lds_b128)
#define HAVE_ASYNC 1
#else
#define HAVE_ASYNC 0
#endif

#if defined(__HIP_DEVICE_COMPILE__) && __has_builtin(__builtin_amdgcn_s_wait_asynccnt)
#define WAIT_ASYNC(n) __builtin_amdgcn_s_wait_asynccnt(n)
#else
#define WAIT_ASYNC(n) asm volatile("s_wait_asynccnt %0" ::"n"(n) : "memory")
#endif

// ============================================================================
// Phase 0: build extended target (blank,l1,blank,l2,...), skip mask, tlen
// ============================================================================
__global__ __launch_bounds__(256) void prep_kernel(const int* __restrict__ ys,
                                                   int* __restrict__ ext,
                                                   int* __restrict__ skip,
                                                   int* __restrict__ tlen) {
  const int b = blockIdx.x;
  __shared__ int tgt[L_];
  __shared__ int cnt;
  if (threadIdx.x == 0) cnt = 0;
  __syncthreads();
  for (int l = threadIdx.x; l < L_; l += blockDim.x) {
    int y = ys[b * L_ + l];
    tgt[l] = (y < 0) ? 0 : y;
    if (y >= 0) atomicAdd(&cnt, 1);
  }
  __syncthreads();
  for (int s = threadIdx.x; s < SPAD; s += blockDim.x) {
    int e = 0, sk = 0;
    if (s < S_ && (s & 1)) {
      int l = (s - 1) >> 1;
      e = tgt[l];
      if (s >= 3) sk = (tgt[l] != tgt[l - 1]) ? 1 : 0;
    }
    ext[b * SPAD + s] = e;
    skip[b * SPAD + s] = sk;
  }
  if (threadIdx.x == 0) tlen[b] = cnt;
}

// ============================================================================
// Phase 1: fused log_softmax + allophone->phoneme logsumexp + ext gather
// One block per (b,t) row. 256 threads, 4 allophones each (C=1024).
// Bandwidth-bound: 196MB of hs_pad in, 49MB em out.
// ============================================================================
__global__ __launch_bounds__(256) void emis_kernel(const float* __restrict__ hs,
                                                   const float* __restrict__ alloW,
                                                   const int* __restrict__ amap,
                                                   const int* __restrict__ ext,
                                                   float* __restrict__ em) {
  const int bt = blockIdx.x;
  const int b = bt / T_;
  const int tid = threadIdx.x;
  const int lane = tid & 31, w = tid >> 5;
  const float* row = hs + (size_t)bt * C_;

  float h[4], sc[4];
  int mapi[4];
  float mh = NEGF, ms = NEGF;
#pragma unroll
  for (int k = 0; k < 4; k++) {
    int idx = tid + k * P_;
    float hv = row[idx];
    float wv = alloW[idx];
    h[k] = hv;
    sc[k] = hv + wv;
    mh = fmaxf(mh, hv);
    ms = fmaxf(ms, sc[k]);
    mapi[k] = amap[idx];
  }

  __shared__ float redA[8], redB[8];
  __shared__ float bc[2];
  __shared__ float bins[P_];
  __shared__ float pe[P_];
  bins[tid] = 0.0f;

  // block max over both mh and ms
  for (int o = 16; o > 0; o >>= 1) {
    mh = fmaxf(mh, __shfl_xor(mh, o));
    ms = fmaxf(ms, __shfl_xor(ms, o));
  }
  if (lane == 0) { redA[w] = mh; redB[w] = ms; }
  __syncthreads();
  if (w == 0) {
    float a = (lane < 8) ? redA[lane] : NEGF;
    float s2 = (lane < 8) ? redB[lane] : NEGF;
    for (int o = 4; o > 0; o >>= 1) {
      a = fmaxf(a, __shfl_xor(a, o));
      s2 = fmaxf(s2, __shfl_xor(s2, o));
    }
    if (lane == 0) { bc[0] = a; bc[1] = s2; }
  }
  __syncthreads();
  const float MH = bc[0], MS = bc[1];

  // sum exp(h-MH) for log_softmax denom; scatter exp(sc-MS) into phoneme bins
  float sh = 0.f;
#pragma unroll
  for (int k = 0; k < 4; k++) {
    sh += __expf(h[k] - MH);
    atomicAdd(&bins[mapi[k]], __expf(sc[k] - MS));  // ds_add_f32
  }
  for (int o = 16; o > 0; o >>= 1) sh += __shfl_xor(sh, o);
  if (lane == 0) redA[w] = sh;
  __syncthreads();
  if (w == 0) {
    float a = (lane < 8) ? redA[lane] : 0.f;
    for (int o = 4; o > 0; o >>= 1) a += __shfl_xor(a, o);
    if (lane == 0) bc[0] = MH + __logf(a);  // row logsumexp
  }
  __syncthreads();
  const float lseH = bc[0];

  float bv = bins[tid];
  pe[tid] = (bv > 0.f) ? (MS + __logf(bv) - lseH) : NEGF;
  __syncthreads();

  // gather phone emissions for the extended target, pad lanes -> NEG
  float* orow = em + (size_t)bt * EMSTR;
  if (tid < SPAD) {
    float v = NEGF;
    if (tid < S_) v = pe[ext[b * SPAD + tid]];
    orow[tid] = v;
  }
}

// ============================================================================
// Phase 2: CTC alpha scan. One wave32 per batch element; alpha lives in
// registers (7 states/lane), neighbor deps via __shfl_up, em rows streamed
// into LDS with double-buffered async-to-LDS copies (ASYNCcnt).
// ============================================================================
__device__ __forceinline__ float lse3f(float x, float y, float z) {
  float m = fmaxf(x, fmaxf(y, z));
  return m + __logf(__expf(x - m) + __expf(y - m) + __expf(z - m));
}

#if HAVE_ASYNC
typedef __attribute__((ext_vector_type(4))) int v4i_t;
typedef __attribute__((address_space(1))) v4i_t* as1_v4i;
typedef __attribute__((address_space(3))) v4i_t* as3_v4i;
__device__ __forceinline__ void copy_row(const float* g, float* l, int lane) {
  // 1KB row = 2 x (32 lanes * 16B) async b128 issues
  as1_v4i g0 = (as1_v4i)(uintptr_t)(g + lane * 4);
  as3_v4i l0 = (as3_v4i)(uint32_t)(uintptr_t)(l + lane * 4);
  __builtin_amdgcn_global_load_async_to_lds_b128(g0, l0, 0, 0);
  as1_v4i g1 = (as1_v4i)(uintptr_t)(g + 128 + lane * 4);
  as3_v4i l1 = (as3_v4i)(uint32_t)(uintptr_t)(l + 128 + lane * 4);
  __builtin_amdgcn_global_load_async_to_lds_b128(g1, l1, 0, 0);
}
#else
__device__ __forceinline__ void copy_row(const float* g, float* l, int lane) {
  const float4* s4 = (const float4*)g;
  float4* d4 = (float4*)l;
  d4[lane] = s4[lane];
  d4[lane + 32] = s4[lane + 32];
}
#endif

__global__ __launch_bounds__(32) void ctc_kernel(const float* __restrict__ em,
                                                 const int* __restrict__ skipArr,
                                                 const int* __restrict__ tlenArr,
                                                 float* __restrict__ loss) {
  __shared__ float stage[2][CHROWS][EMSTR];
  __shared__ float fin[2];
  const int b = blockIdx.x;
  const int lane = threadIdx.x;
  const float* emb = em + (size_t)b * T_ * EMSTR;

  float a[7];
  int sk[7];
#pragma unroll
  for (int j = 0; j < 7; j++) {
    int s = lane * 7 + j;
    sk[j] = skipArr[b * SPAD + s];
    a[j] = (s < 2) ? emb[s] : NEGF;  // alpha0
  }

  constexpr int NROWS = T_ - 1;                          // 1499
  constexpr int NC = (NROWS + CHROWS - 1) / CHROWS;      // 188
  constexpr int LASTROWS = NROWS - (NC - 1) * CHROWS;    // 3

  // prime chunk 0
  {
    const int rows0 = (NC == 1) ? LASTROWS : CHROWS;
    for (int r = 0; r < rows0; r++)
      copy_row(emb + (size_t)(1 + r) * EMSTR, &stage[0][r][0], lane);
  }

  for (int c = 0; c < NC; c++) {
    const int rows = (c == NC - 1) ? LASTROWS : CHROWS;
    if (c + 1 < NC) {
      const int nrows = (c + 1 == NC - 1) ? LASTROWS : CHROWS;
      for (int r = 0; r < nrows; r++)
        copy_row(emb + (size_t)(1 + (c + 1) * CHROWS + r) * EMSTR,
                 &stage[(c + 1) & 1][r][0], lane);
    }
#if HAVE_ASYNC
    // async loads complete in order: leaving <= next-chunk-issues outstanding
    // guarantees the current chunk's data has landed in LDS.
    if (c + 1 < NC) {
      if (c + 1 == NC - 1) { WAIT_ASYNC(2 * LASTROWS); }
      else                 { WAIT_ASYNC(2 * CHROWS); }
    } else {
      WAIT_ASYNC(0);
    }
    asm volatile("" ::: "memory");
#endif
    for (int r = 0; r < rows; r++) {
      const float* erow = &stage[c & 1][r][0];
      float p6 = __shfl_up(a[6], 1);  // prev lane's alpha[s-1] for our j=0
      float p5 = __shfl_up(a[5], 1);  // prev lane's alpha[s-2] for our j=0
      if (lane == 0) { p6 = NEGF; p5 = NEGF; }
      float n[7];
#pragma unroll
      for (int j = 0; j < 7; j++) {
        float a1 = (j >= 1) ? a[j - 1] : p6;
        float a2 = (j >= 2) ? a[j - 2] : ((j == 1) ? p6 : p5);
        a2 = sk[j] ? a2 : NEGF;
        n[j] = erow[lane * 7 + j] + lse3f(a[j], a1, a2);
      }
#pragma unroll
      for (int j = 0; j < 7; j++) a[j] = n[j];
    }
  }

  const int tl = tlenArr[b];
  const int sl = 2 * tl, sp = 2 * tl - 1;
#pragma unroll
  for (int j = 0; j < 7; j++) {
    int s = lane * 7 + j;
    if (s == sl) fin[0] = a[j];
    if (s == sp) fin[1] = a[j];
  }
  __syncthreads();
  if (lane == 0) {
    float x = fin[0], y = fin[1];
    float m = fmaxf(x, y);
    loss[b] = -(m + __logf(__expf(x - m) + __expf(y - m)));
  }
}

// ============================================================================
// Phase 3: filtered mean over batch
// ============================================================================
__global__ __launch_bounds__(32) void final_kernel(const float* __restrict__ loss,
                                                   float* __restrict__ out) {
  const int lane = threadIdx.x;
  float l = loss[lane];
  l = (fabsf(l) < 1e29f) ? l : 0.0f;  // NaN/inf/huge -> 0 (NaN compares false)
  for (int o = 16; o > 0; o >>= 1) l += __shfl_xor(l, o);
  if (lane == 0) out[0] = l / (float)B_;
}

// ============================================================================
extern "C" void kernel_launch(void* const* d_in, const int* in_sizes, int n_in,
                              void* d_out, int out_size, void* d_ws, size_t ws_size,
                              hipStream_t stream) {
  (void)in_sizes; (void)n_in; (void)out_size; (void)ws_size;
  const float* alloW = (const float*)d_in[0];
  const float* hs    = (const float*)d_in[1];
  // d_in[2] = hlens (unused by the reference math)
  const int* ys      = (const int*)d_in[3];
  const int* amap    = (const int*)d_in[4];

  char* ws = (char*)d_ws;
  float* em = (float*)ws;
  size_t off = (size_t)B_ * T_ * EMSTR * sizeof(float);
  int* ext  = (int*)(ws + off); off += (size_t)B_ * SPAD * sizeof(int);
  int* skip = (int*)(ws + off); off += (size_t)B_ * SPAD * sizeof(int);
  int* tlen = (int*)(ws + off); off += (size_t)B_ * sizeof(int);
  float* loss = (float*)(ws + off);

  prep_kernel<<<B_, 256, 0, stream>>>(ys, ext, skip, tlen);
  emis_kernel<<<B_ * T_, 256, 0, stream>>>(hs, alloW, amap, ext, em);
  ctc_kernel<<<B_, 32, 0, stream>>>(em, skip, tlen, loss);
  final_kernel<<<1, 32, 0, stream>>>(loss, (float*)d_out);
}